// RoiAlignExtractor_1202590843770
// MI455X (gfx1250) — compile-verified
//
#include <hip/hip_runtime.h>
#include <math.h>

// FPN RoI-Align for MI455X (gfx1250).
// Strategy: per-ROI rectangular feature sub-region is DMA'd into LDS with the
// Tensor Data Mover (3D tile: x=region_w, y=region_h, z=8 channels; row stride
// W, plane stride H*W), double-buffered across 32 channel tiles and tracked
// with TENSORcnt. All bilinear taps then read LDS instead of doing scattered
// per-lane global gathers. Level math / sample prep matches the JAX reference.

#define OUTS    7
#define NSAMP   14            // OUTS * SR (SR == 2)
#define CT      8             // channels per LDS tile
#define NCH     256
#define NTILE   (NCH / CT)    // 32
#define THREADS 256
#define MAXC    1088          // max region cells per channel (bound: < ~960)

typedef unsigned int tdm_v4u __attribute__((ext_vector_type(4)));
typedef int          tdm_v8i __attribute__((ext_vector_type(8)));
typedef int          tdm_v4i __attribute__((ext_vector_type(4)));

#if __has_builtin(__builtin_amdgcn_tensor_load_to_lds)
#define HAVE_TDM 1
#else
#define HAVE_TDM 0
#endif

__device__ __forceinline__ int rfl(int x) { return __builtin_amdgcn_readfirstlane(x); }

__global__ __launch_bounds__(THREADS, 2)
void roi_align_fpn_kernel(const float* __restrict__ f0, const float* __restrict__ f1,
                          const float* __restrict__ f2, const float* __restrict__ f3,
                          const float* __restrict__ rois, float* __restrict__ out)
{
    __shared__ float s_tile[2][CT * MAXC];
    __shared__ int   s_lo[2 * NSAMP], s_hi[2 * NSAMP];
    __shared__ float s_fr[2 * NSAMP], s_va[2 * NSAMP];
    __shared__ int   s_meta[6]; // xmin, ymin, rw, rh, lvl, batch

    const int n   = blockIdx.x;
    const int tid = threadIdx.x;

    // ---- Per-ROI metadata (separable x/y sample descriptors + region bounds)
    if (tid == 0) {
        const float* r = rois + (size_t)n * 5;
        float scale = sqrtf(fmaxf((r[3] - r[1]) * (r[4] - r[2]), 0.0f));
        int lvl = (int)floorf(log2f(scale * (1.0f / 56.0f) + 1e-6f));
        lvl = lvl < 0 ? 0 : (lvl > 3 ? 3 : lvl);
        const int W = (lvl == 0) ? 336 : ((lvl == 1) ? 168 : ((lvl == 2) ? 84 : 42));
        const int H = (lvl == 0) ? 200 : ((lvl == 1) ? 100 : ((lvl == 2) ? 50 : 25));
        const float ss = 1.0f / (float)(4 << lvl);
        const float px1 = r[1] * ss - 0.5f, px2 = r[3] * ss - 0.5f;
        const float py1 = r[2] * ss - 0.5f, py2 = r[4] * ss - 0.5f;
        const float bw = (px2 - px1) * (1.0f / 7.0f);
        const float bh = (py2 - py1) * (1.0f / 7.0f);
        int xmin = W - 1, xmax = 0, ymin = H - 1, ymax = 0;
        for (int i = 0; i < NSAMP; ++i) {
            const float offc = (float)(i >> 1) + (float)(i & 1) * 0.5f + 0.25f;
            // x axis
            float c  = px1 + offc * bw;
            float va = (c >= -1.0f && c <= (float)W) ? 1.0f : 0.0f;
            float cc = fminf(fmaxf(c, 0.0f), (float)W - 1.0f);
            int lo = (int)floorf(cc);
            int hi = min(lo + 1, W - 1);
            s_lo[i] = lo; s_hi[i] = hi; s_fr[i] = cc - (float)lo; s_va[i] = va;
            xmin = min(xmin, lo); xmax = max(xmax, hi);
            // y axis
            c  = py1 + offc * bh;
            va = (c >= -1.0f && c <= (float)H) ? 1.0f : 0.0f;
            cc = fminf(fmaxf(c, 0.0f), (float)H - 1.0f);
            lo = (int)floorf(cc);
            hi = min(lo + 1, H - 1);
            s_lo[NSAMP + i] = lo; s_hi[NSAMP + i] = hi;
            s_fr[NSAMP + i] = cc - (float)lo; s_va[NSAMP + i] = va;
            ymin = min(ymin, lo); ymax = max(ymax, hi);
        }
        s_meta[0] = xmin; s_meta[1] = ymin;
        s_meta[2] = xmax - xmin + 1; s_meta[3] = ymax - ymin + 1;
        s_meta[4] = lvl; s_meta[5] = (int)r[0];
    }
    __syncthreads();

    const int xmin = s_meta[0], ymin = s_meta[1];
    const int rw = s_meta[2], rh = s_meta[3];
    const int lvl = s_meta[4], b = s_meta[5];
    const int W = (lvl == 0) ? 336 : ((lvl == 1) ? 168 : ((lvl == 2) ? 84 : 42));
    const int H = (lvl == 0) ? 200 : ((lvl == 1) ? 100 : ((lvl == 2) ? 50 : 25));
    const float* fb = (lvl == 0) ? f0 : ((lvl == 1) ? f1 : ((lvl == 2) ? f2 : f3));
    const int cells = rw * rh;

    // ---- Per-thread output descriptors (channel-invariant; hoisted out of tile loop)
    int    coff[2][16];
    float  cwgt[2][16];
    int    cbase[2];
    size_t obase[2];
#pragma unroll
    for (int o = 0; o < 2; ++o) {
        const int idx = tid + o * THREADS;
        if (idx < CT * 49) {
            const int ci = idx / 49;
            const int p  = idx - ci * 49;
            const int ph = p / 7, pw = p - ph * 7;
            cbase[o] = ci * cells;
            obase[o] = ((size_t)n * NCH + ci) * 49 + p;
#pragma unroll
            for (int s = 0; s < 4; ++s) {
                const int sy = s >> 1, sx = s & 1;
                const int ix = pw * 2 + sx;
                const int iy = NSAMP + ph * 2 + sy;
                const int x0r = s_lo[ix] - xmin, x1r = s_hi[ix] - xmin;
                const int y0r = s_lo[iy] - ymin, y1r = s_hi[iy] - ymin;
                const float fx = s_fr[ix], fy = s_fr[iy];
                const float va = s_va[ix] * s_va[iy];
                coff[o][s * 4 + 0] = y0r * rw + x0r;
                coff[o][s * 4 + 1] = y0r * rw + x1r;
                coff[o][s * 4 + 2] = y1r * rw + x0r;
                coff[o][s * 4 + 3] = y1r * rw + x1r;
                cwgt[o][s * 4 + 0] = va * (1.0f - fy) * (1.0f - fx);
                cwgt[o][s * 4 + 1] = va * (1.0f - fy) * fx;
                cwgt[o][s * 4 + 2] = va * fy * (1.0f - fx);
                cwgt[o][s * 4 + 3] = va * fy * fx;
            }
        }
    }

    // ---- Tile loader: TDM 3D-tile DMA (wave0 issues) or cooperative fallback
    auto tile_load = [&](int ct, int buf) {
#if HAVE_TDM
        if (tid < 32) {  // wave 0 only (uniform per wave); TDM ignores EXEC
            const int c0 = ct * CT;
            const unsigned long long ga = (unsigned long long)(const void*)
                (fb + (((size_t)b * NCH + c0) * H + ymin) * (size_t)W + xmin);
            const unsigned ldsb = (unsigned)(size_t)&s_tile[buf][0];
            const unsigned td0 = (unsigned)(W - xmin);   // tensor extent dim0 (x)
            const unsigned td1 = (unsigned)(H - ymin);   // tensor extent dim1 (y)
            const unsigned hw  = (unsigned)(H * W);      // dim1 (plane/z) stride
            tdm_v4u g0; tdm_v8i g1; tdm_v4i g2, g3; tdm_v8i g4;
            // D# group0: count=1 | lds_addr | global_addr[56:0] | type=2
            g0[0] = (unsigned)rfl(1);
            g0[1] = (unsigned)rfl((int)ldsb);
            g0[2] = (unsigned)rfl((int)(unsigned)ga);
            g0[3] = (unsigned)rfl((int)((unsigned)(ga >> 32) | (2u << 30)));
            // D# group1: data_size=4B; tensor_dim0/1; tile_dim0/1/2; strides
            g1[0] = rfl((int)(2u << 16));
            g1[1] = rfl((int)((td0 & 0xFFFFu) << 16));
            g1[2] = rfl((int)((td0 >> 16) | ((td1 & 0xFFFFu) << 16)));
            g1[3] = rfl((int)((td1 >> 16) | ((unsigned)rw << 16)));
            g1[4] = rfl((int)((unsigned)rh | ((unsigned)CT << 16)));
            g1[5] = rfl((int)(unsigned)W);               // dim0_stride (row) lo32
            g1[6] = rfl((int)((hw & 0xFFFFu) << 16));    // dim0_stride hi16=0 | dim1_stride lo16
            g1[7] = rfl((int)(hw >> 16));                // dim1_stride [47:16]
            // D# group2: tensor_dim2 (z extent for OOB); no dim3, no iterate
            g2[0] = rfl((int)(NCH - c0));
            g2[1] = 0; g2[2] = 0; g2[3] = 0;
            g3[0] = 0; g3[1] = 0; g3[2] = 0; g3[3] = 0;
            // extra operand of the 6-arg (clang-23) builtin form: zero-filled
            g4[0] = 0; g4[1] = 0; g4[2] = 0; g4[3] = 0;
            g4[4] = 0; g4[5] = 0; g4[6] = 0; g4[7] = 0;
            __builtin_amdgcn_tensor_load_to_lds(g0, g1, g2, g3, g4, 0);
        }
#else
        const int c0 = ct * CT;
        const size_t sb = ((size_t)b * NCH + c0) * (size_t)(H * W);
        const int total = CT * cells;
        for (int i = tid; i < total; i += THREADS) {
            const int z  = i / cells;
            const int rm = i - z * cells;
            const int yy = rm / rw;
            const int xx = rm - yy * rw;
            s_tile[buf][i] = fb[sb + (size_t)z * (H * W) + (size_t)(ymin + yy) * W + (xmin + xx)];
        }
#endif
    };

    // ---- Double-buffered channel-tile pipeline
    tile_load(0, 0);
    for (int ct = 0; ct < NTILE; ++ct) {
        const int buf = ct & 1;
        if (ct + 1 < NTILE) tile_load(ct + 1, buf ^ 1);
#if HAVE_TDM
        if (tid < 32) {
            if (ct + 1 < NTILE) __builtin_amdgcn_s_wait_tensorcnt(1); // tile ct done, ct+1 in flight
            else                __builtin_amdgcn_s_wait_tensorcnt(0);
        }
#endif
        __syncthreads();   // tile ct visible to all waves

        const float* Ft = &s_tile[buf][0];
#pragma unroll
        for (int o = 0; o < 2; ++o) {
            const int idx = tid + o * THREADS;
            if (idx < CT * 49) {
                float acc = 0.0f;
#pragma unroll
                for (int k = 0; k < 16; ++k)
                    acc += cwgt[o][k] * Ft[cbase[o] + coff[o][k]];
                out[obase[o] + (size_t)ct * (CT * 49)] = acc * 0.25f;
            }
        }
        __syncthreads();   // all reads of this buffer done before it is re-filled
    }
}

extern "C" void kernel_launch(void* const* d_in, const int* in_sizes, int n_in,
                              void* d_out, int out_size, void* d_ws, size_t ws_size,
                              hipStream_t stream) {
    (void)n_in; (void)out_size; (void)d_ws; (void)ws_size;
    const float* f0   = (const float*)d_in[0];
    const float* f1   = (const float*)d_in[1];
    const float* f2   = (const float*)d_in[2];
    const float* f3   = (const float*)d_in[3];
    const float* rois = (const float*)d_in[4];
    const int N = in_sizes[4] / 5;   // 512 ROIs
    roi_align_fpn_kernel<<<N, THREADS, 0, stream>>>(f0, f1, f2, f3, rois, (float*)d_out);
}